// FuzzyMultiLayer_20547123544801
// MI455X (gfx1250) — compile-verified
//
#include <hip/hip_runtime.h>
#include <math.h>

typedef __attribute__((ext_vector_type(2))) float v2f;
typedef __attribute__((ext_vector_type(4))) float v4f;
typedef __attribute__((ext_vector_type(8))) float v8f;

#define B_DIM 8
#define H_DIM 256
#define W_DIM 256
#define C_DIM 8
#define K_DIM 16
#define HW    (H_DIM * W_DIM)          // 65536
#define NPTS  (B_DIM * HW)             // 524288
#define OUTC  (C_DIM + K_DIM * C_DIM)  // 136
#define KSTRIDE 80                     // floats per k in ws: 64 Linv + 8 t + 1 logdet + pad
#define WS_FLOATS (K_DIM * KSTRIDE)    // 1280
#define KC_STRIDE 557056u              // 8*512*136 : output-float stride per (k*8+c) step

// ---------------------------------------------------------------------------
// Precompute per-k: Linv = L^-1 (lower-tri), t = -Linv*mu, logdet = sum log|Lii|
// ---------------------------------------------------------------------------
__global__ void fml_precompute(const float* __restrict__ scale,
                               const float* __restrict__ mean,
                               float* __restrict__ ws) {
    int k = threadIdx.x;
    if (k >= K_DIM) return;
    const float* L  = scale + k * C_DIM * C_DIM;
    const float* mu = mean  + k * C_DIM;

    float Linv[C_DIM][C_DIM];
    #pragma unroll
    for (int i = 0; i < C_DIM; i++)
        #pragma unroll
        for (int j = 0; j < C_DIM; j++) Linv[i][j] = 0.0f;

    #pragma unroll
    for (int j = 0; j < C_DIM; j++) {
        Linv[j][j] = 1.0f / L[j * C_DIM + j];
        #pragma unroll
        for (int i = 0; i < C_DIM; i++) {
            if (i > j) {
                float s = 0.0f;
                for (int m = j; m < i; m++) s += L[i * C_DIM + m] * Linv[m][j];
                Linv[i][j] = -s / L[i * C_DIM + i];
            }
        }
    }

    float* w = ws + k * KSTRIDE;
    #pragma unroll
    for (int i = 0; i < C_DIM; i++)
        #pragma unroll
        for (int j = 0; j < C_DIM; j++) w[i * C_DIM + j] = Linv[i][j];

    #pragma unroll
    for (int i = 0; i < C_DIM; i++) {
        float s = 0.0f;
        for (int j = 0; j <= i; j++) s += Linv[i][j] * mu[j];
        w[64 + i] = -s;                        // t = -Linv * mu
    }
    float logdet = 0.0f;
    #pragma unroll
    for (int i = 0; i < C_DIM; i++) logdet += logf(fabsf(L[i * C_DIM + i]));
    w[72] = logdet;
}

// ---------------------------------------------------------------------------
// Main kernel: one wave = one tile of 16 consecutive spatial points; the
// k-pair loop stacks [Linv_k0 ; Linv_k1] into the 16 WMMA rows so every
// v_wmma_f32_16x16x4_f32 lane-slot does useful work.
// All indices are 32-bit (output = 71.3M floats < 2^31) so stores use the
// SGPR-base + 32-bit VGPR-offset form instead of per-lane 64-bit address math.
// ---------------------------------------------------------------------------
__global__ __launch_bounds__(256) void fml_main(const float* __restrict__ x,
                                                const float* __restrict__ ws,
                                                float* __restrict__ out) {
    __shared__ float lws[WS_FLOATS];           // 5 KB: Linv/t/logdet for all 16 k
    #pragma unroll
    for (int i = 0; i < WS_FLOATS / 256; i++)
        lws[threadIdx.x + 256 * i] = ws[threadIdx.x + 256 * i];
    __syncthreads();

    const int lane = threadIdx.x & 31;
    const int wave = threadIdx.x >> 5;
    const unsigned tile = blockIdx.x * 8u + wave;     // 16-point tile id
    const unsigned s0   = tile * 16u;                 // first global point of tile
    const unsigned bb   = s0 >> 16;                   // s0 / HW
    const unsigned p0   = s0 & (HW - 1);              // s0 % HW (multiple of 16)

    const int n    = lane & 15;                       // point within tile
    const int half = lane >> 4;                       // wave half (0/1)

    // lane's own x row (point n): 32-bit indexing, contiguous 32B
    const float* xp = x + (s0 + n) * (unsigned)C_DIM;
    float xr[8];
    #pragma unroll
    for (int c = 0; c < 8; c++) xr[c] = xp[c];

    // B operands (k-invariant): lane l, vgpr v -> N=l%16, K = v + 2*(l/16)
    v2f b1, b2;
    b1.x = xr[2 * half + 0];  b1.y = xr[2 * half + 1];   // channels 0..3
    b2.x = xr[2 * half + 4];  b2.y = xr[2 * half + 5];   // channels 4..7

    // x pass-through: final[s*136 + c] = x[s*8 + c]
    if (half == 0) {
        float* o = out + (s0 + n) * (unsigned)OUTC;
        *(v4f*)(o + 0) = *(const v4f*)(xr + 0);
        *(v4f*)(o + 4) = *(const v4f*)(xr + 4);
    }

    // out index for (k,c,point n):  obase + (k*8+c)*KC_STRIDE   (all 32-bit)
    const unsigned obase = (bb * 512u + (p0 >> 7)) * 136u + 8u + (p0 & 127u) + (unsigned)n;

    const float NORM = 7.3515082656373775f;    // 0.5 * C * log(2*pi)
    const int  kA  = (n >> 3) & 1;             // which k of the pair feeds A rows
    const int  row = lane & 7;                 // Linv row for A operand

    #pragma unroll 2
    for (int kp = 0; kp < 8; kp++) {
        const int k = 2 * kp + half;           // this lane's k for D rows / gauss

        // A operand from LDS: lane l, vgpr v -> M=l%16, K = v + 2*(l/16)
        const float* wa = lws + (2 * kp + kA) * KSTRIDE + row * 8 + 2 * half;
        const v2f a1 = *(const v2f*)(wa + 0);  // cols (0,1)/(2,3) by half
        const v2f a2 = *(const v2f*)(wa + 4);  // cols (4,5)/(6,7) by half

        // C operand: t_k broadcast across columns (C/D: M = v + 8*half)
        const float* wt = lws + k * KSTRIDE + 64;
        const v4f t0 = *(const v4f*)(wt + 0);
        const v4f t1 = *(const v4f*)(wt + 4);
        v8f acc;
        acc[0] = t0.x; acc[1] = t0.y; acc[2] = t0.z; acc[3] = t0.w;
        acc[4] = t1.x; acc[5] = t1.y; acc[6] = t1.z; acc[7] = t1.w;

        acc = __builtin_amdgcn_wmma_f32_16x16x4_f32(false, a1, false, b1,
                                                    (short)0, acc, false, false);
        acc = __builtin_amdgcn_wmma_f32_16x16x4_f32(false, a2, false, b2,
                                                    (short)0, acc, false, false);

        // quad = ||y||^2 : lane holds the full y-vector for (k, point n)
        float quad = 0.0f;
        #pragma unroll
        for (int v = 0; v < 8; v++) quad += acc[v] * acc[v];

        const float logdet = lws[k * KSTRIDE + 72];
        const float g = __expf(-0.5f * quad - logdet - NORM);

        // 16 contiguous floats per (k,c) per lane-half; 32-bit voffsets
        const unsigned kbase = obase + (unsigned)(k * 8) * KC_STRIDE;
        #pragma unroll
        for (int c = 0; c < 8; c++)
            out[kbase + (unsigned)c * KC_STRIDE] = g * xr[c];
    }
}

extern "C" void kernel_launch(void* const* d_in, const int* in_sizes, int n_in,
                              void* d_out, int out_size, void* d_ws, size_t ws_size,
                              hipStream_t stream) {
    const float* x     = (const float*)d_in[0];
    const float* scale = (const float*)d_in[1];
    const float* mean  = (const float*)d_in[2];
    float* out = (float*)d_out;
    float* ws  = (float*)d_ws;   // needs 16*80*4 = 5120 bytes

    fml_precompute<<<1, 32, 0, stream>>>(scale, mean, ws);

    const int tiles  = NPTS / 16;       // 32768
    const int blocks = tiles / 8;       // 4096 (8 waves/block, wave32)
    fml_main<<<blocks, 256, 0, stream>>>(x, ws, out);
}